// CrossAttentionQuerySelector_21337397526909
// MI455X (gfx1250) — compile-verified
//
#include <hip/hip_runtime.h>

#define DD 256
#define HH 8
#define HDD 32
#define SS 3
#define KK 7
#define TB 16          // tokens per workgroup
#define NWAVE 8        // 8 wave32 = 256 threads

#define KV_ROWS (TB*KK)   // 112
#define SQ_ROWS (TB*SS)   // 48
#define KVS 260           // padded stride (floats) for 256-wide LDS panels
#define QS  260
#define H1S 516           // padded stride for 512-wide h1 panel

typedef __attribute__((ext_vector_type(2))) float v2f;
typedef __attribute__((ext_vector_type(8))) float v8f;

// Explicit global (addrspace(1)) pointer so cand-selected / weight streams lower to
// global_load_b64 instead of flat_load_b64 (flat burns DScnt + the LDS pipe).
typedef const float __attribute__((address_space(1)))* gcf;
static __device__ __forceinline__ gcf as_global(const float* p) {
  return (gcf)(unsigned long long)p;
}

// One 16x16 f32 output tile, accumulating over Ksize with V_WMMA_F32_16X16X4_F32.
// aRow/bRow are per-lane pointers with the lane's kOff (=(lane>>4)*2) folded in:
//   A(m,k): lane = m (0..15 per half), k = kOff + j
//   B(k,n): lane = n,                  k = kOff + j
template <typename PA, typename PB>
static __device__ __forceinline__ v8f wmma_kloop(PA aRow, PB bRow, int Ksize, v8f c) {
#pragma unroll 16
  for (int kb = 0; kb < Ksize; kb += 4) {
    v2f a, b;
    a.x = aRow[kb];     a.y = aRow[kb + 1];
    b.x = bRow[kb];     b.y = bRow[kb + 1];
    c = __builtin_amdgcn_wmma_f32_16x16x4_f32(false, a, false, b, (short)0, c,
                                              false, false);
  }
  return c;
}

extern "C" __global__ void __launch_bounds__(256, 1)
caqs_fused(const float* __restrict__ c0, const float* __restrict__ c1,
           const float* __restrict__ c2, const float* __restrict__ c3,
           const float* __restrict__ c4, const float* __restrict__ c5,
           const float* __restrict__ c6,
           const float* __restrict__ slotq,
           const float* __restrict__ ipw,  const float* __restrict__ ipb,
           const float* __restrict__ outw, const float* __restrict__ outb,
           const float* __restrict__ ln1g, const float* __restrict__ ln1b,
           const float* __restrict__ w1,   const float* __restrict__ b1,
           const float* __restrict__ w2,   const float* __restrict__ b2,
           const float* __restrict__ ln2g, const float* __restrict__ ln2b,
           float* __restrict__ out)
{
  __shared__ float sm[2*KV_ROWS*KVS + SQ_ROWS*QS + SS*DD + 2*SQ_ROWS];
  float* kh    = sm;                      // 112x260  (later reused as qpre)
  float* vh    = kh + KV_ROWS*KVS;        // 112x260  (later reused as h1, stride 516)
  float* qb    = vh + KV_ROWS*KVS;        // 48x260   ctx -> q
  float* qhs   = qb + SQ_ROWS*QS;         // 3x256 scaled slot queries
  float* mstat = qhs + SS*DD;             // 48 means
  float* rstat = mstat + SQ_ROWS;         // 48 rsigmas
  float* qpre  = kh;                      // alias: pre-LN rows (48x260)
  float* h1    = vh;                      // alias: gelu output (48x516)

  const int tid  = threadIdx.x;
  const int wave = tid >> 5;
  const int lane = tid & 31;
  const int l16  = lane & 15;
  const int half = lane >> 4;
  const int kOff = half * 2;
  const int tok0 = blockIdx.x * TB;

  const float* cand[KK] = {c0, c1, c2, c3, c4, c5, c6};
  const v8f cz = {0.f, 0.f, 0.f, 0.f, 0.f, 0.f, 0.f, 0.f};

  // ---------- Phase 0: qh = (slotq @ wq^T + bq)/sqrt(HD) via WMMA.
  // One M-tile: rows 0..2 are the slots, rows 3..15 computed on duplicated A rows
  // and discarded. 16 N-tiles over 8 waves.
  for (int nt = wave; nt < 16; nt += NWAVE) {
    const int col = nt*16 + l16;
    gcf aRow = as_global(slotq + (l16 % SS)*DD + kOff);
    gcf bRow = as_global(ipw + (size_t)col * DD + kOff);   // wq rows 0..255
    v8f c = wmma_kloop(aRow, bRow, DD, cz);
    if (half == 0) {
      const float bias = ipb[col];
#pragma unroll
      for (int r = 0; r < SS; ++r)
        qhs[r*DD + col] = (c[r] + bias) * 0.17677669529663687f;  // 1/sqrt(32)
    }
  }

  // ---------- Phase 1: KV projection: (112 x 256) @ (256 x 512) -> kh | vh in LDS
  for (int tile = wave; tile < 7*32; tile += NWAVE) {
    const int mt = tile % 7, nt = tile / 7;
    const int row = mt*16 + l16;                      // 0..111, row = t*7 + kcand
    const int t  = row / KK, kc = row % KK;
    gcf aRow = as_global(cand[kc] + (size_t)(tok0 + t) * DD + kOff);
    const int col = nt*16 + l16;                      // 0..511 (0..255 kh, 256..511 vh)
    gcf bRow = as_global(ipw + (size_t)(DD + col) * DD + kOff); // wk rows 256.., wv 512..
    v8f c = wmma_kloop(aRow, bRow, DD, cz);
    const float bias = ipb[DD + col];
    float* dbuf = (col < DD) ? kh : vh;
    const int cc = col & (DD - 1);
    const int rb = mt*16 + 8*half;
#pragma unroll
    for (int r = 0; r < 8; ++r) dbuf[(rb + r)*KVS + cc] = c[r] + bias;
  }
  __syncthreads();

  // ---------- Phase 2: attention (one (token, head) per thread; tid < 128)
  if (tid < TB*HH) {
    const int t = tid >> 3, h = tid & 7;
    float sc[SS][KK];
#pragma unroll
    for (int s = 0; s < SS; ++s) {
#pragma unroll
      for (int kc = 0; kc < KK; ++kc) {
        const float* kr = kh + (t*KK + kc)*KVS + h*HDD;
        const float* qr = qhs + s*DD + h*HDD;
        float a = 0.f;
#pragma unroll
        for (int d = 0; d < HDD; ++d) a += qr[d] * kr[d];
        sc[s][kc] = a;
      }
      float mx = sc[s][0];
#pragma unroll
      for (int kc = 1; kc < KK; ++kc) mx = fmaxf(mx, sc[s][kc]);
      float sum = 0.f;
#pragma unroll
      for (int kc = 0; kc < KK; ++kc) { float e = expf(sc[s][kc] - mx); sc[s][kc] = e; sum += e; }
      const float inv = 1.0f / sum;
#pragma unroll
      for (int d = 0; d < HDD; ++d) {
        float a = 0.f;
#pragma unroll
        for (int kc = 0; kc < KK; ++kc) a += sc[s][kc] * vh[(t*KK + kc)*KVS + h*HDD + d];
        qb[(t*SS + s)*QS + h*HDD + d] = a * inv;      // ctx
      }
    }
  }
  __syncthreads();

  // ---------- Phase 3: attn_out = ctx @ out_w^T + out_b + slotq  -> qpre (kh region)
  for (int tile = wave; tile < 3*16; tile += NWAVE) {
    const int mt = tile % 3, nt = tile / 3;
    const float* aRow = qb + (mt*16 + l16)*QS + kOff;           // LDS
    const int col = nt*16 + l16;
    gcf bRow = as_global(outw + (size_t)col * DD + kOff);
    v8f c = wmma_kloop(aRow, bRow, DD, cz);
    const float bias = outb[col];
    const int rb = mt*16 + 8*half;
#pragma unroll
    for (int r = 0; r < 8; ++r) {
      const int row = rb + r;                         // row = t*3 + s
      qpre[row*QS + col] = c[r] + bias + slotq[(row % SS)*DD + col];
    }
  }
  __syncthreads();

  // ---------- LN1: qpre -> qb
  if (tid < SQ_ROWS) {
    const float* x = qpre + tid*QS;
    float m = 0.f;
    for (int d = 0; d < DD; ++d) m += x[d];
    m *= (1.0f/DD);
    float v = 0.f;
    for (int d = 0; d < DD; ++d) { float u = x[d] - m; v += u*u; }
    v *= (1.0f/DD);
    mstat[tid] = m;
    rstat[tid] = rsqrtf(v + 1e-5f);
  }
  __syncthreads();
  {
    const float g = ln1g[tid], b = ln1b[tid];
    for (int j = 0; j < SQ_ROWS; ++j)
      qb[j*QS + tid] = (qpre[j*QS + tid] - mstat[j]) * rstat[j] * g + b;
  }
  __syncthreads();

  // ---------- MLP1: gelu(q @ w1^T + b1) -> h1 (vh region, stride 516)
  for (int tile = wave; tile < 3*32; tile += NWAVE) {
    const int mt = tile % 3, nt = tile / 3;
    const float* aRow = qb + (mt*16 + l16)*QS + kOff;           // LDS
    const int col = nt*16 + l16;                      // 0..511
    gcf bRow = as_global(w1 + (size_t)col * DD + kOff);
    v8f c = wmma_kloop(aRow, bRow, DD, cz);
    const float bias = b1[col];
    const int rb = mt*16 + 8*half;
#pragma unroll
    for (int r = 0; r < 8; ++r) {
      const float x = c[r] + bias;
      h1[(rb + r)*H1S + col] = 0.5f * x * (1.0f + erff(x * 0.70710678118654752f));
    }
  }
  __syncthreads();

  // ---------- MLP2: h1 @ w2^T + b2 + q(residual) -> qpre (kh region)
  for (int tile = wave; tile < 3*16; tile += NWAVE) {
    const int mt = tile % 3, nt = tile / 3;
    const float* aRow = h1 + (mt*16 + l16)*H1S + kOff;          // LDS, K=512
    const int col = nt*16 + l16;
    gcf bRow = as_global(w2 + (size_t)col * (2*DD) + kOff);
    v8f c = wmma_kloop(aRow, bRow, 2*DD, cz);
    const float bias = b2[col];
    const int rb = mt*16 + 8*half;
#pragma unroll
    for (int r = 0; r < 8; ++r) {
      const int row = rb + r;
      qpre[row*QS + col] = c[r] + bias + qb[row*QS + col];
    }
  }
  __syncthreads();

  // ---------- LN2 -> global out (coalesced: one column per thread)
  if (tid < SQ_ROWS) {
    const float* x = qpre + tid*QS;
    float m = 0.f;
    for (int d = 0; d < DD; ++d) m += x[d];
    m *= (1.0f/DD);
    float v = 0.f;
    for (int d = 0; d < DD; ++d) { float u = x[d] - m; v += u*u; }
    v *= (1.0f/DD);
    mstat[tid] = m;
    rstat[tid] = rsqrtf(v + 1e-5f);
  }
  __syncthreads();
  {
    const float g = ln2g[tid], b = ln2b[tid];
    const size_t base = (size_t)blockIdx.x * SQ_ROWS * DD;
    for (int j = 0; j < SQ_ROWS; ++j)
      out[base + (size_t)j*DD + tid] =
          (qpre[j*QS + tid] - mstat[j]) * rstat[j] * g + b;
  }
}

extern "C" void kernel_launch(void* const* d_in, const int* in_sizes, int n_in,
                              void* d_out, int out_size, void* d_ws, size_t ws_size,
                              hipStream_t stream) {
  (void)n_in; (void)d_ws; (void)ws_size; (void)out_size;
  const float* c0   = (const float*)d_in[0];
  const float* c1   = (const float*)d_in[1];
  const float* c2   = (const float*)d_in[2];
  const float* c3   = (const float*)d_in[3];
  const float* c4   = (const float*)d_in[4];
  const float* c5   = (const float*)d_in[5];
  const float* c6   = (const float*)d_in[6];
  const float* sq   = (const float*)d_in[7];
  const float* ipw  = (const float*)d_in[8];
  const float* ipb  = (const float*)d_in[9];
  const float* outw = (const float*)d_in[10];
  const float* outb = (const float*)d_in[11];
  const float* l1g  = (const float*)d_in[12];
  const float* l1b  = (const float*)d_in[13];
  const float* w1   = (const float*)d_in[14];
  const float* b1   = (const float*)d_in[15];
  const float* w2   = (const float*)d_in[16];
  const float* b2   = (const float*)d_in[17];
  const float* l2g  = (const float*)d_in[18];
  const float* l2b  = (const float*)d_in[19];
  float* out        = (float*)d_out;

  const int nTok   = in_sizes[0] / DD;   // B*T = 32768
  const int blocks = nTok / TB;          // 2048
  hipLaunchKernelGGL(caqs_fused, dim3(blocks), dim3(256), 0, stream,
                     c0, c1, c2, c3, c4, c5, c6, sq, ipw, ipb, outw, outb,
                     l1g, l1b, w1, b1, w2, b2, l2g, l2b, out);
}